// ChargeFieldModule_41394894799546
// MI455X (gfx1250) — compile-verified
//
#include <hip/hip_runtime.h>

typedef __attribute__((ext_vector_type(2))) float v2f;
typedef __attribute__((ext_vector_type(4))) float v4f;
typedef __attribute__((ext_vector_type(8))) float v8f;

#define GRID_N   256
#define K_CHUNK  2048   // rects per (supertile, chunk) wave; 512 steps x 4 WMMAs

__global__ __launch_bounds__(256) void cf_zero(float* out, int n) {
    int i = blockIdx.x * 256 + threadIdx.x;
    if (i < n) out[i] = 0.0f;
}

// One K=4 rank-update step feeding a 2x2 block of 16x16 output tiles.
// Loads + rect edges computed once, reused by 2 A-pairs and 2 B-pairs -> 4 WMMAs.
#define CF_STEP(R, D, W0, W1)                                                          \
    {                                                                                  \
        float bx1a = (R).x + (D).x, by1a = (R).y + (D).y;                              \
        float bx1b = (R).z + (D).z, by1b = (R).w + (D).w;                              \
        v2f A0 = { fmaxf(fminf(gx1_0, bx1a) - fmaxf(gx0_0, (R).x), 0.f) * (W0),        \
                   fmaxf(fminf(gx1_0, bx1b) - fmaxf(gx0_0, (R).z), 0.f) * (W1) };      \
        v2f A1 = { fmaxf(fminf(gx1_1, bx1a) - fmaxf(gx0_1, (R).x), 0.f) * (W0),        \
                   fmaxf(fminf(gx1_1, bx1b) - fmaxf(gx0_1, (R).z), 0.f) * (W1) };      \
        v2f B0 = { fmaxf(fminf(gy1_0, by1a) - fmaxf(gy0_0, (R).y), 0.f),               \
                   fmaxf(fminf(gy1_0, by1b) - fmaxf(gy0_0, (R).w), 0.f) };             \
        v2f B1 = { fmaxf(fminf(gy1_1, by1a) - fmaxf(gy0_1, (R).y), 0.f),               \
                   fmaxf(fminf(gy1_1, by1b) - fmaxf(gy0_1, (R).w), 0.f) };             \
        c00 = __builtin_amdgcn_wmma_f32_16x16x4_f32(false, A0, false, B0, (short)0, c00, false, false); \
        c01 = __builtin_amdgcn_wmma_f32_16x16x4_f32(false, A0, false, B1, (short)0, c01, false, false); \
        c10 = __builtin_amdgcn_wmma_f32_16x16x4_f32(false, A1, false, B0, (short)0, c10, false, false); \
        c11 = __builtin_amdgcn_wmma_f32_16x16x4_f32(false, A1, false, B1, (short)0, c11, false, false); \
    }

__global__ __launch_bounds__(256) void cf_wmma(
    const float* __restrict__ boundary,   // [xmin, ymin, xmax, ymax]
    const float* __restrict__ xy,         // N x 2
    const float* __restrict__ dims,       // N x 2
    const float* __restrict__ w,          // N
    float* __restrict__ out,              // 256 x 256
    int nrects)
{
    const int lane = threadIdx.x & 31;
    const int wave = threadIdx.x >> 5;            // 0..7
    const int st   = blockIdx.y * 8 + wave;       // supertile 0..63 (2x2 tiles each)
    const int ti0  = (st >> 3) * 2;               // row tiles ti0, ti0+1
    const int tj0  = (st & 7) * 2;                 // col tiles tj0, tj0+1
    const int half = lane >> 4;                   // 0: WMMA K=0,1   1: K=2,3
    const int lp   = lane & 15;                   // M (A) / N (B) within a tile

    const float xmin = boundary[0], ymin = boundary[1];
    const float xmax = boundary[2], ymax = boundary[3];
    const float lx  = (xmax - xmin) * (1.0f / GRID_N);
    const float ly  = (ymax - ymin) * (1.0f / GRID_N);
    const float inv = 1.0f / (lx * ly);

    // Loop-invariant per-lane grid-cell intervals for both row tiles / col tiles.
    const float gx0_0 = xmin + (float)(ti0 * 16 + lp) * lx;
    const float gx1_0 = gx0_0 + lx;
    const float gx0_1 = gx0_0 + 16.0f * lx;
    const float gx1_1 = gx0_1 + lx;
    const float gy0_0 = ymin + (float)(tj0 * 16 + lp) * ly;
    const float gy1_0 = gy0_0 + ly;
    const float gy0_1 = gy0_0 + 16.0f * ly;
    const float gy1_1 = gy0_1 + ly;

    v8f c00 = {0.f,0.f,0.f,0.f,0.f,0.f,0.f,0.f};
    v8f c01 = c00, c10 = c00, c11 = c00;

    const int kbeg = blockIdx.x * K_CHUNK;

    if (kbeg + K_CHUNK <= nrects) {
        // Fast path: whole chunk in bounds -> no clamps, no predication,
        // pointer+immediate-offset addressing.
        const float* pxy = xy   + 2 * (kbeg + 2 * half);
        const float* pdm = dims + 2 * (kbeg + 2 * half);
        const float* pw  = w    + (kbeg + 2 * half);
        #pragma unroll 4
        for (int kb = 0; kb < K_CHUNK; kb += 4) {
            v4f r  = *(const v4f*)(pxy + 2 * kb);   // x,y of rects k0, k0+1
            v4f d  = *(const v4f*)(pdm + 2 * kb);
            v2f ww = *(const v2f*)(pw + kb);
            float w0 = ww.x * inv;
            float w1 = ww.y * inv;
            CF_STEP(r, d, w0, w1);
        }
    } else {
        // Tail chunk: clamped addresses, zeroed weights kill OOB contributions
        // (A column of zeros zeroes the rank-1 update regardless of B garbage).
        const int kend = nrects;
        for (int kb = kbeg; kb < kend; kb += 4) {
            int k0  = kb + 2 * half;
            int k0c = min(k0, nrects - 2);
            v4f r  = *(const v4f*)(xy   + 2 * k0c);
            v4f d  = *(const v4f*)(dims + 2 * k0c);
            v2f ww = *(const v2f*)(w + k0c);
            float w0 = (k0     < nrects) ? ww.x * inv : 0.0f;
            float w1 = (k0 + 1 < nrects) ? ww.y * inv : 0.0f;
            CF_STEP(r, d, w0, w1);
        }
    }

    // C/D layout: VGPR v holds M = v + 8*half, N = lp.
    const int j0  = tj0 * 16 + lp;
    const int j1  = j0 + 16;
    const int ib0 = ti0 * 16 + half * 8;
    const int ib1 = ib0 + 16;
    #pragma unroll
    for (int v = 0; v < 8; ++v) {
        atomicAdd(out + (ib0 + v) * GRID_N + j0, c00[v]);
        atomicAdd(out + (ib0 + v) * GRID_N + j1, c01[v]);
        atomicAdd(out + (ib1 + v) * GRID_N + j0, c10[v]);
        atomicAdd(out + (ib1 + v) * GRID_N + j1, c11[v]);
    }
}

extern "C" void kernel_launch(void* const* d_in, const int* in_sizes, int n_in,
                              void* d_out, int out_size, void* d_ws, size_t ws_size,
                              hipStream_t stream) {
    const float* boundary = (const float*)d_in[0];
    const float* xy       = (const float*)d_in[1];
    const float* dims     = (const float*)d_in[2];
    const float* w        = (const float*)d_in[3];
    float* out = (float*)d_out;
    const int nrects = in_sizes[3];            // charge_weight element count

    cf_zero<<<dim3((out_size + 255) / 256), 256, 0, stream>>>(out, out_size);

    const int chunks = (nrects + K_CHUNK - 1) / K_CHUNK;
    dim3 grid(chunks, 8);                      // 8 blocks x 8 waves = 64 supertiles
    cf_wmma<<<grid, 256, 0, stream>>>(boundary, xy, dims, w, out, nrects);
}